// SmplBody25FaceLayer_26834955665851
// MI455X (gfx1250) — compile-verified
//
#include <hip/hip_runtime.h>
#include <hip/hip_bf16.h>

#define NBATCH 512
#define NV     6890
#define NJ     24
#define NB     10
#define NP     207
#define C3     (NV*3)        // 20670
#define C3PAD  20672         // C3 rounded up to multiple of 16
#define NK     95            // 25 body + 70 face
#define NKPAD  96
#define KPAD   6892          // NV rounded up to multiple of 4
#define PFW    208           // padded pose-feature width (52 * 4)
#define NCOLS  (NBATCH*3)    // 1536

typedef float v2f __attribute__((ext_vector_type(2)));
typedef float v8f __attribute__((ext_vector_type(8)));

// ---------------- Stage 0a: zero-padded copy of posedirs (PFW x C3PAD) ----------------
__global__ void k_padposedirs(const float* __restrict__ posedirs, float* __restrict__ pd) {
    size_t idx = (size_t)blockIdx.x * blockDim.x + threadIdx.x;
    if (idx >= (size_t)PFW * C3PAD) return;
    int r = (int)(idx / C3PAD);
    int c = (int)(idx % C3PAD);
    pd[idx] = (r < NP && c < C3) ? posedirs[(size_t)r * C3 + c] : 0.f;
}

// ---------------- Stage 0b: concatenated zero-padded regressors (NKPAD x KPAD) ----------------
__global__ void k_padreg(const float* __restrict__ body25, const float* __restrict__ face,
                         float* __restrict__ regcat) {
    size_t idx = (size_t)blockIdx.x * blockDim.x + threadIdx.x;
    if (idx >= (size_t)NKPAD * KPAD) return;
    int m = (int)(idx / KPAD);
    int k = (int)(idx % KPAD);
    float val = 0.f;
    if (m < NK && k < NV)
        val = (m < 25) ? body25[(size_t)m * NV + k] : face[(size_t)(m - 25) * NV + k];
    regcat[idx] = val;
}

// ---------------- Stage 0c: zero the 2 K-pad rows of vertsT ----------------
__global__ void k_zpadverts(float* __restrict__ vertsT) {
    int idx = blockIdx.x * blockDim.x + threadIdx.x;
    if (idx < (KPAD - NV) * NCOLS) vertsT[(size_t)NV * NCOLS + idx] = 0.f;
}

// ---------------- Stage 1: v_ss = betas @ shapedirs + v_template ----------------
__global__ void k_shape(const float* __restrict__ betas, const float* __restrict__ shapedirs,
                        const float* __restrict__ v_template, float* __restrict__ v_ss) {
    __shared__ float bs[NB];
    int n = blockIdx.y;
    if (threadIdx.x < NB) bs[threadIdx.x] = betas[n * NB + threadIdx.x];
    __syncthreads();
    int v = blockIdx.x * blockDim.x + threadIdx.x;
    if (v >= NV) return;
    for (int c = 0; c < 3; ++c) {
        int col = v * 3 + c;
        float acc = v_template[col];
        for (int b = 0; b < NB; ++b) acc += bs[b] * shapedirs[(size_t)b * C3 + col];
        v_ss[(size_t)n * C3 + col] = acc;
    }
}

// ---------------- Stage 2: per-batch scale from body height ----------------
__global__ void k_scale(const float* __restrict__ v_ss, float* __restrict__ scale) {
    int n = blockIdx.x * blockDim.x + threadIdx.x;
    if (n >= NBATCH) return;
    const float* vs = v_ss + (size_t)n * C3;
    float h = vs[2802 * 3 + 1] + vs[6262 * 3 + 1] - (vs[2237 * 3 + 1] + vs[6728 * 3 + 1]);
    scale[n] = 1.66f / h;
}

// ---------------- Stage 3: J = scale * (J_regressor @ v_ss) ----------------
__global__ void k_joints(const float* __restrict__ Jreg, const float* __restrict__ v_ss,
                         const float* __restrict__ scale, float* __restrict__ Jout) {
    int j = blockIdx.x, n = blockIdx.y, tid = threadIdx.x;
    const float* vs = v_ss + (size_t)n * C3;
    const float* jr = Jreg + (size_t)j * NV;
    float a0 = 0.f, a1 = 0.f, a2 = 0.f;
    for (int v = tid; v < NV; v += 256) {
        float w = jr[v];
        a0 += w * vs[v * 3 + 0];
        a1 += w * vs[v * 3 + 1];
        a2 += w * vs[v * 3 + 2];
    }
    __shared__ float s0[256], s1[256], s2[256];
    s0[tid] = a0; s1[tid] = a1; s2[tid] = a2;
    __syncthreads();
    for (int off = 128; off > 0; off >>= 1) {
        if (tid < off) { s0[tid] += s0[tid + off]; s1[tid] += s1[tid + off]; s2[tid] += s2[tid + off]; }
        __syncthreads();
    }
    if (tid == 0) {
        float sc = scale[n];
        float* o = Jout + ((size_t)n * NJ + j) * 3;
        o[0] = sc * s0[0]; o[1] = sc * s1[0]; o[2] = sc * s2[0];
    }
}

// ---------------- Stage 4a: pose_feature (padded to 208, last col = 0) ----------------
__global__ void k_posefeat(const float* __restrict__ pose, float* __restrict__ pf) {
    int idx = blockIdx.x * blockDim.x + threadIdx.x;
    if (idx >= NBATCH * PFW) return;
    int n = idx / PFW, p = idx % PFW;
    float val = 0.f;
    if (p < NP) {
        int j = 1 + p / 9;
        int r = (p % 9) / 3;
        int c = p % 3;
        val = pose[(((size_t)n * NJ + j) * 3 + r) * 3 + c] - ((r == c) ? 1.0f : 0.0f);
    }
    pf[idx] = val;
}

// ---- Stage 4b: v_posed = pf @ posedirs_pad + scale*v_ss (WMMA f32, branch-free loop) ----
__global__ void k_posed(const float* __restrict__ pf, const float* __restrict__ pd,
                        const float* __restrict__ scale, float* __restrict__ v_sp) {
    int lane  = threadIdx.x;
    int tileC = blockIdx.x;   // 0..(C3PAD/16 - 1)
    int tileM = blockIdx.y;   // 0..31
    int row = lane & 15;
    int hi  = lane >> 4;      // 0 or 1
    int col = tileC * 16 + row;
    const float* arow = pf + (size_t)(tileM * 16 + row) * PFW;
    const float* bcol = pd + col;
    v8f acc = {};
    for (int kk = 0; kk < 52; ++kk) {
        int k0 = kk * 4 + hi * 2;
        v2f a, b;
        a.x = arow[k0];
        a.y = arow[k0 + 1];
        b.x = bcol[(size_t)(k0)     * C3PAD];
        b.y = bcol[(size_t)(k0 + 1) * C3PAD];
        __builtin_prefetch(&bcol[(size_t)(k0 + 8) * C3PAD], 0, 1);
        acc = __builtin_amdgcn_wmma_f32_16x16x4_f32(false, a, false, b, (short)0, acc, false, false);
    }
    if (col < C3) {
        for (int i = 0; i < 8; ++i) {
            int m = tileM * 16 + i + hi * 8;
            size_t o = (size_t)m * C3 + col;
            v_sp[o] = acc[i] + scale[m] * v_sp[o];
        }
    }
}

// ---------------- Stage 5: forward kinematics chain ----------------
__global__ void k_fk(const float* __restrict__ pose, const float* __restrict__ Jbuf,
                     const int* __restrict__ parents, float* __restrict__ res,
                     float* __restrict__ Afin) {
    int n = blockIdx.x * blockDim.x + threadIdx.x;
    if (n >= NBATCH) return;
    const float* Jn = Jbuf + (size_t)n * NJ * 3;
    for (int j = 0; j < NJ; ++j) {
        float R[9];
        const float* pr = pose + ((size_t)n * NJ + j) * 9;
        for (int e = 0; e < 9; ++e) R[e] = pr[e];
        float t[3];
        float Rg[9], tg[3];
        if (j == 0) {
            for (int c = 0; c < 3; ++c) t[c] = Jn[c];
            for (int e = 0; e < 9; ++e) Rg[e] = R[e];
            for (int c = 0; c < 3; ++c) tg[c] = t[c];
        } else {
            int p = parents[j];
            for (int c = 0; c < 3; ++c) t[c] = Jn[j * 3 + c] - Jn[p * 3 + c];
            const float* pp = res + ((size_t)n * NJ + p) * 12;  // rows of [R|t]
            for (int r = 0; r < 3; ++r) {
                for (int c = 0; c < 3; ++c)
                    Rg[r * 3 + c] = pp[r * 4 + 0] * R[0 * 3 + c] +
                                    pp[r * 4 + 1] * R[1 * 3 + c] +
                                    pp[r * 4 + 2] * R[2 * 3 + c];
                tg[r] = pp[r * 4 + 0] * t[0] + pp[r * 4 + 1] * t[1] +
                        pp[r * 4 + 2] * t[2] + pp[r * 4 + 3];
            }
        }
        float* rr = res + ((size_t)n * NJ + j) * 12;
        for (int r = 0; r < 3; ++r) {
            rr[r * 4 + 0] = Rg[r * 3 + 0];
            rr[r * 4 + 1] = Rg[r * 3 + 1];
            rr[r * 4 + 2] = Rg[r * 3 + 2];
            rr[r * 4 + 3] = tg[r];
        }
        float jx = Jn[j * 3 + 0], jy = Jn[j * 3 + 1], jz = Jn[j * 3 + 2];
        float* aa = Afin + ((size_t)n * NJ + j) * 12;
        for (int r = 0; r < 3; ++r) {
            aa[r * 4 + 0] = Rg[r * 3 + 0];
            aa[r * 4 + 1] = Rg[r * 3 + 1];
            aa[r * 4 + 2] = Rg[r * 3 + 2];
            aa[r * 4 + 3] = tg[r] - (Rg[r * 3 + 0] * jx + Rg[r * 3 + 1] * jy + Rg[r * 3 + 2] * jz);
        }
    }
}

// ---------------- Stage 6: LBS skinning; writes verts transposed (KPAD, N*3) ----------------
__global__ void k_lbs(const float* __restrict__ Afin, const float* __restrict__ lbs_w,
                      const float* __restrict__ v_posed, const float* __restrict__ trans,
                      float* __restrict__ vertsT) {
    __shared__ float As[NJ * 12];
    int n = blockIdx.y;
    for (int i = threadIdx.x; i < NJ * 12; i += blockDim.x)
        As[i] = Afin[(size_t)n * NJ * 12 + i];
    __syncthreads();
    int v = blockIdx.x * blockDim.x + threadIdx.x;
    if (v >= NV) return;
    float T[12];
    for (int e = 0; e < 12; ++e) T[e] = 0.f;
    const float* wv = lbs_w + (size_t)v * NJ;
    for (int j = 0; j < NJ; ++j) {
        float w = wv[j];
        const float* aj = As + j * 12;
        for (int e = 0; e < 12; ++e) T[e] += w * aj[e];
    }
    const float* p = v_posed + (size_t)n * C3 + v * 3;
    float x = p[0], y = p[1], z = p[2];
    float* o = vertsT + ((size_t)v * NBATCH + n) * 3;
    for (int r = 0; r < 3; ++r)
        o[r] = T[r * 4 + 0] * x + T[r * 4 + 1] * y + T[r * 4 + 2] * z + T[r * 4 + 3] + trans[n * 3 + r];
}

// ---- Stage 7: keypoint regression GEMM (WMMA f32, branch-free): out = regcat @ vertsT ----
__global__ void k_reg(const float* __restrict__ regcat, const float* __restrict__ vertsT,
                      float* __restrict__ out) {
    int lane  = threadIdx.x;
    int tileC = blockIdx.x;   // 0..95
    int tileM = blockIdx.y;   // 0..5
    int row = lane & 15;
    int hi  = lane >> 4;
    int m   = tileM * 16 + row;
    int col = tileC * 16 + row;
    const float* arow = regcat + (size_t)m * KPAD;
    const float* bcol = vertsT + col;
    v8f acc = {};
    for (int kk = 0; kk < KPAD / 4; ++kk) {
        int k0 = kk * 4 + hi * 2;
        v2f a, b;
        a.x = arow[k0];
        a.y = arow[k0 + 1];
        b.x = bcol[(size_t)(k0)     * NCOLS];
        b.y = bcol[(size_t)(k0 + 1) * NCOLS];
        __builtin_prefetch(&bcol[(size_t)(k0 + 16) * NCOLS], 0, 1);
        acc = __builtin_amdgcn_wmma_f32_16x16x4_f32(false, a, false, b, (short)0, acc, false, false);
    }
    for (int i = 0; i < 8; ++i) {
        int mo = tileM * 16 + i + hi * 8;
        if (mo < NK) {
            int n = col / 3, c = col % 3;
            out[((size_t)n * NK + mo) * 3 + c] = acc[i];
        }
    }
}

extern "C" void kernel_launch(void* const* d_in, const int* in_sizes, int n_in,
                              void* d_out, int out_size, void* d_ws, size_t ws_size,
                              hipStream_t stream) {
    (void)in_sizes; (void)n_in; (void)out_size; (void)ws_size;
    const float* pose       = (const float*)d_in[0];
    const float* betas      = (const float*)d_in[1];
    const float* trans      = (const float*)d_in[2];
    const float* v_template = (const float*)d_in[3];
    const float* shapedirs  = (const float*)d_in[4];
    const float* J_reg      = (const float*)d_in[5];
    const float* posedirs   = (const float*)d_in[6];
    const float* lbs_w      = (const float*)d_in[7];
    const float* body25     = (const float*)d_in[8];
    const float* face       = (const float*)d_in[9];
    const int*   parents    = (const int*)d_in[10];
    float* out = (float*)d_out;

    // workspace carve-up (floats)
    float* w      = (float*)d_ws;
    float* v_ss   = w; w += (size_t)NBATCH * C3;        // overwritten in place by v_posed
    float* scale  = w; w += NBATCH;
    float* Jbuf   = w; w += (size_t)NBATCH * NJ * 3;
    float* pf     = w; w += (size_t)NBATCH * PFW;
    float* res    = w; w += (size_t)NBATCH * NJ * 12;
    float* Afin   = w; w += (size_t)NBATCH * NJ * 12;
    float* vertsT = w; w += (size_t)KPAD * NCOLS;       // K-padded, transposed verts
    float* pdpad  = w; w += (size_t)PFW * C3PAD;        // padded posedirs
    float* regcat = w; w += (size_t)NKPAD * KPAD;       // padded concat regressors

    k_padposedirs<<<(int)(((size_t)PFW * C3PAD + 255) / 256), 256, 0, stream>>>(posedirs, pdpad);
    k_padreg<<<(int)(((size_t)NKPAD * KPAD + 255) / 256), 256, 0, stream>>>(body25, face, regcat);
    k_zpadverts<<<((KPAD - NV) * NCOLS + 255) / 256, 256, 0, stream>>>(vertsT);

    k_shape<<<dim3((NV + 255) / 256, NBATCH), 256, 0, stream>>>(betas, shapedirs, v_template, v_ss);
    k_scale<<<(NBATCH + 255) / 256, 256, 0, stream>>>(v_ss, scale);
    k_joints<<<dim3(NJ, NBATCH), 256, 0, stream>>>(J_reg, v_ss, scale, Jbuf);
    k_posefeat<<<(NBATCH * PFW + 255) / 256, 256, 0, stream>>>(pose, pf);
    k_posed<<<dim3(C3PAD / 16, NBATCH / 16), 32, 0, stream>>>(pf, pdpad, scale, v_ss);
    k_fk<<<(NBATCH + 63) / 64, 64, 0, stream>>>(pose, Jbuf, parents, res, Afin);
    k_lbs<<<dim3((NV + 127) / 128, NBATCH), 128, 0, stream>>>(Afin, lbs_w, v_ss, trans, vertsT);
    k_reg<<<dim3(NCOLS / 16, 6), 32, 0, stream>>>(regcat, vertsT, out);
}